// GOT_71159018160346
// MI455X (gfx1250) — compile-verified
//
#include <hip/hip_runtime.h>

typedef __attribute__((ext_vector_type(2))) float v2f;
typedef __attribute__((ext_vector_type(8))) float v8f;

#define NN 4096
#define LOG2E 1.4426950408889634f
#define LN2   0.6931471805599453f
#define NEG_INF (-3.402823466e38f)

// A0 = (mean + std*eps) / TAU, and zero the C offset vector.
__global__ __launch_bounds__(256) void sk_init(const float* __restrict__ eps,
                                               const float* __restrict__ mean,
                                               const float* __restrict__ stdv,
                                               float* __restrict__ A0,
                                               float* __restrict__ Cvec) {
    size_t idx = (size_t)blockIdx.x * 256 + threadIdx.x;
    A0[idx] = (mean[idx] + stdv[idx] * eps[idx]) * 0.5f;  // TAU = 2.0
    if (idx < NN) Cvec[idx] = 0.0f;
}

// Row pass: R[i] = logsumexp_j(A0[i,j] - C[j]).
// One block = 16 rows. 8 waves, each wave owns a 512-column K slice.
// Row sums computed with V_WMMA_F32_16X16X4_F32 against an all-ones B.
// Both per-wave loops have a constant trip count (128) so the compiler
// emits scalar loops: EXEC stays all-1s around every WMMA by construction.
__global__ __launch_bounds__(256) void sk_row(const float* __restrict__ A0,
                                              const float* __restrict__ Cvec,
                                              float* __restrict__ Rvec) {
    __shared__ float sC[NN];        // 16 KB staged copy of C
    __shared__ float sMax[8 * 32];  // per (wave,lane) partial max
    __shared__ float sM[16];        // per-row max
    __shared__ float sSum[8 * 16];  // per-wave row partial sums

    const int tid  = threadIdx.x;
    const int wave = tid >> 5;
    const int lane = tid & 31;
    const int hi   = lane >> 4;   // 0: K%4 in {0,1}; 1: K%4 in {2,3}
    const int row  = lane & 15;
    const int r0   = blockIdx.x * 16;

    for (int k = tid; k < NN; k += 256) sC[k] = Cvec[k];
    __syncthreads();

    const float* Arow = A0 + (size_t)(r0 + row) * NN;
    const int kBase = wave * 512 + 2 * hi;  // lane's first column in its slice

    // Phase 1: per-row max of (A0 - C) over this wave's K slice.
    float m = NEG_INF;
    for (int t = 0; t < 128; ++t) {
        const int k = kBase + 4 * t;
        v2f a = *(const v2f*)(Arow + k);
        float x0 = a.x - sC[k];
        float x1 = a.y - sC[k + 1];
        m = fmaxf(m, fmaxf(x0, x1));
    }
    sMax[wave * 32 + lane] = m;
    __syncthreads();
    if (tid < 16) {
        float mm = NEG_INF;
        for (int w = 0; w < 8; ++w) {
            mm = fmaxf(mm, sMax[w * 32 + tid]);
            mm = fmaxf(mm, sMax[w * 32 + 16 + tid]);
        }
        sM[tid] = mm;
    }
    __syncthreads();
    const float mrow = sM[row];

    // Phase 2: row sums of exp(A0 - C - m) via WMMA f32 16x16x4, B = ones.
    v8f acc = {};
    v2f ones;
    ones.x = 1.0f;
    ones.y = 1.0f;
    for (int t = 0; t < 128; ++t) {
        const int k = kBase + 4 * t;
        v2f a = *(const v2f*)(Arow + k);
        v2f e;
        e.x = exp2f((a.x - sC[k]     - mrow) * LOG2E);
        e.y = exp2f((a.y - sC[k + 1] - mrow) * LOG2E);
        acc = __builtin_amdgcn_wmma_f32_16x16x4_f32(
            /*neg_a=*/false, e, /*neg_b=*/false, ones,
            /*c_mod=*/(short)0, acc, /*reuse_a=*/false, /*reuse_b=*/false);
    }
    // D columns are identical; lanes with N==0 (lanes 0 and 16) hold rows
    // v + 8*hi in acc[v].
    if ((lane & 15) == 0) {
        for (int v = 0; v < 8; ++v) sSum[wave * 16 + hi * 8 + v] = acc[v];
    }
    __syncthreads();
    if (tid < 16) {
        float s = 0.0f;
        for (int w = 0; w < 8; ++w) s += sSum[w * 16 + tid];
        Rvec[r0 + tid] = sM[tid] + log2f(s) * LN2;
    }
}

// Column pass stage A: partial (max, scaled-sum) over a 256-row slice.
// grid = (16 col-blocks, 16 row-slices), consecutive threads = consecutive
// columns -> fully coalesced.
__global__ __launch_bounds__(256) void sk_col_partial(const float* __restrict__ A0,
                                                      const float* __restrict__ Rvec,
                                                      float* __restrict__ mPart,
                                                      float* __restrict__ sPart) {
    const int j  = blockIdx.x * 256 + threadIdx.x;
    const int i0 = blockIdx.y * 256;
    const float* p = A0 + (size_t)i0 * NN + j;

    float m = NEG_INF;
    for (int i = 0; i < 256; ++i) {
        __builtin_prefetch(p + (size_t)(i + 8) * NN, 0, 0);
        float x = p[(size_t)i * NN] - Rvec[i0 + i];
        m = fmaxf(m, x);
    }
    float s = 0.0f;
    for (int i = 0; i < 256; ++i) {
        float x = p[(size_t)i * NN] - Rvec[i0 + i];
        s += exp2f((x - m) * LOG2E);
    }
    mPart[blockIdx.y * NN + j] = m;
    sPart[blockIdx.y * NN + j] = s;
}

// Column pass stage B: combine 16 (m,s) partials per column, LSE-style.
__global__ __launch_bounds__(256) void sk_col_combine(const float* __restrict__ mPart,
                                                      const float* __restrict__ sPart,
                                                      float* __restrict__ Cvec) {
    const int j = blockIdx.x * 256 + threadIdx.x;
    float m = NEG_INF;
    for (int p = 0; p < 16; ++p) m = fmaxf(m, mPart[p * NN + j]);
    float s = 0.0f;
    for (int p = 0; p < 16; ++p)
        s += sPart[p * NN + j] * exp2f((mPart[p * NN + j] - m) * LOG2E);
    Cvec[j] = m + log2f(s) * LN2;
}

// out = exp(A0 - R[i] - C[j]), in place on d_out.
__global__ __launch_bounds__(256) void sk_final(float* __restrict__ A0,
                                                const float* __restrict__ Rvec,
                                                const float* __restrict__ Cvec) {
    size_t idx = (size_t)blockIdx.x * 256 + threadIdx.x;
    const int i = (int)(idx >> 12);
    const int j = (int)(idx & (NN - 1));
    A0[idx] = exp2f((A0[idx] - Rvec[i] - Cvec[j]) * LOG2E);
}

extern "C" void kernel_launch(void* const* d_in, const int* in_sizes, int n_in,
                              void* d_out, int out_size, void* d_ws, size_t ws_size,
                              hipStream_t stream) {
    const float* eps  = (const float*)d_in[0];
    const float* mean = (const float*)d_in[1];
    const float* stdv = (const float*)d_in[2];

    float* A0 = (float*)d_out;        // 4096x4096 log-domain matrix, exp'd in place at end
    float* ws = (float*)d_ws;
    float* Rvec  = ws;                 // 4096
    float* Cvec  = ws + NN;            // 4096
    float* mPart = ws + 2 * NN;        // 16*4096
    float* sPart = ws + 2 * NN + 16 * NN;

    const int elemBlocks = (NN * NN) / 256;  // 65536

    sk_init<<<elemBlocks, 256, 0, stream>>>(eps, mean, stdv, A0, Cvec);

    for (int it = 0; it < 30; ++it) {
        sk_row<<<NN / 16, 256, 0, stream>>>(A0, Cvec, Rvec);
        sk_col_partial<<<dim3(16, 16), 256, 0, stream>>>(A0, Rvec, mPart, sPart);
        sk_col_combine<<<NN / 256, 256, 0, stream>>>(mPart, sPart, Cvec);
    }

    sk_final<<<elemBlocks, 256, 0, stream>>>(A0, Rvec, Cvec);
}